// GraphSAGE_PI_2173253452589
// MI455X (gfx1250) — compile-verified
//
#include <hip/hip_runtime.h>
#include <hip/hip_bf16.h>

// ---------------------------------------------------------------------------
// GraphSAGE (3x SAGEConv mean-aggr + linear head) for gfx1250 (MI455X).
// Dense GEMM parts use V_WMMA_F32_16X16X4_F32 (full fp32 precision, wave32).
// ---------------------------------------------------------------------------

#define N_NODES 50000
#define N_EDGES 600000
#define IN_DIM  16
#define HIDDEN  128
#define OUT_DIM 4

typedef __attribute__((ext_vector_type(2))) float v2f;
typedef __attribute__((ext_vector_type(8))) float v8f;

static_assert(N_NODES % 16 == 0, "row tiles assumed full");

// ---------------------------------------------------------------------------
// Utility kernels
// ---------------------------------------------------------------------------
__global__ void zero_kernel(float* __restrict__ p, long long n) {
    long long i = (long long)blockIdx.x * blockDim.x + threadIdx.x;
    if (i < n) p[i] = 0.0f;
}

__global__ void count_kernel(const long long* __restrict__ dst,
                             float* __restrict__ cnt, int nEdges) {
    int e = blockIdx.x * blockDim.x + threadIdx.x;
    if (e < nEdges) atomicAdd(&cnt[(int)dst[e]], 1.0f);
}

__global__ void inv_kernel(const float* __restrict__ cnt,
                           float* __restrict__ inv, int n) {
    int i = blockIdx.x * blockDim.x + threadIdx.x;
    if (i < n) inv[i] = 1.0f / fmaxf(cnt[i], 1.0f);
}

// ---------------------------------------------------------------------------
// Edge scatter: s[dst] += x[src]  (float4 chunks, consecutive lanes cover one
// edge's feature row -> coalesced gather + coalesced global_atomic_add_f32)
// ---------------------------------------------------------------------------
template <int F>
__global__ void scatter_add_kernel(const float* __restrict__ x,
                                   const long long* __restrict__ src,
                                   const long long* __restrict__ dst,
                                   float* __restrict__ s, int nEdges) {
    constexpr int CPE = F / 4;  // float4 chunks per edge
    long long gid = (long long)blockIdx.x * blockDim.x + threadIdx.x;
    long long total = (long long)nEdges * CPE;
    if (gid >= total) return;
    int e = (int)(gid / CPE);
    int c = (int)(gid % CPE);
    int sr = (int)src[e];
    int dr = (int)dst[e];
    float4 v = *(const float4*)(x + (size_t)sr * F + c * 4);
    float* p = s + (size_t)dr * F + c * 4;
    atomicAdd(p + 0, v.x);
    atomicAdd(p + 1, v.y);
    atomicAdd(p + 2, v.z);
    atomicAdd(p + 3, v.w);
}

// ---------------------------------------------------------------------------
// Fused SAGE GEMM:  out = relu?( (s * inv) @ Wl^T + x @ Wr^T + b )
// Wave32 computes a 16x16 output tile with V_WMMA_F32_16X16X4_F32.
//   A lane L:  A[M = L%16][k0 + 2*(L/16) .. +1]   (float2)
//   B lane L:  W[n = L%16][k0 + 2*(L/16) .. +1]   (float2, row-major weights)
//   D vgpr v:  row = v + 8*(L/16), col = L%16
// ---------------------------------------------------------------------------
template <int K, int NOUT, bool RELU, bool HAS_L>
__global__ __launch_bounds__(256) void sage_gemm_kernel(
    const float* __restrict__ s, const float* __restrict__ inv,
    const float* __restrict__ x, const float* __restrict__ Wl,
    const float* __restrict__ Wr, const float* __restrict__ bias,
    float* __restrict__ out, int nrows) {
    constexpr int NT = (NOUT + 15) / 16;  // output-col tiles
    constexpr int RT = 8 / NT;            // row tiles per block (8 waves)
    constexpr int LSTR = K + 4;           // padded LDS row stride (floats)
    constexpr bool USE_LDS = HAS_L;       // head (single term) reads global
    constexpr bool FULLCOL = (NOUT % 16 == 0);  // all 16 lanes map to columns

    __shared__ float ldsX[USE_LDS ? RT * 16 * LSTR : 1];
    __shared__ float ldsS[USE_LDS ? RT * 16 * LSTR : 1];

    const int tid = threadIdx.x;
    const int blockRow0 = blockIdx.x * (RT * 16);

    if constexpr (USE_LDS) {
        // Cooperative stage of A tiles (x and inv-scaled s) into LDS.
        constexpr int CH = RT * 16 * (K / 4);  // float4 chunks per buffer
        for (int c = tid; c < CH; c += 256) {
            int row = c / (K / 4);
            int kk = (c % (K / 4)) * 4;
            int grow = blockRow0 + row;
            float4 vx = make_float4(0.f, 0.f, 0.f, 0.f);
            float4 vs = make_float4(0.f, 0.f, 0.f, 0.f);
            if (grow < nrows) {
                vx = *(const float4*)(x + (size_t)grow * K + kk);
                float sc = inv[grow];
                float4 t = *(const float4*)(s + (size_t)grow * K + kk);
                vs = make_float4(t.x * sc, t.y * sc, t.z * sc, t.w * sc);
            }
            *(float4*)(ldsX + row * LSTR + kk) = vx;  // LSTR%4==0 -> aligned
            *(float4*)(ldsS + row * LSTR + kk) = vs;
        }
        __syncthreads();
    }

    const int wave = tid >> 5;
    const int lane = tid & 31;
    const int l16 = lane & 15;
    const int lh = lane >> 4;
    const int rowTileInBlk = wave / NT;
    const int nTile = wave % NT;
    const int rowBase = blockRow0 + rowTileInBlk * 16;
    if (rowBase >= nrows) return;  // wave-uniform exit: EXEC stays all-ones
    const int col = nTile * 16 + l16;
    const bool colValid = FULLCOL ? true : (col < NOUT);

    v8f acc = {0.f, 0.f, 0.f, 0.f, 0.f, 0.f, 0.f, 0.f};

    if constexpr (HAS_L) {
        const float* lsr = ldsS + (rowTileInBlk * 16 + l16) * LSTR;
        const float* wrow = Wl + (size_t)(colValid ? col : 0) * K;
#pragma unroll 4
        for (int k0 = 0; k0 < K; k0 += 4) {
            int k = k0 + 2 * lh;
            v2f a = *(const v2f*)(lsr + k);
            v2f b = colValid ? *(const v2f*)(wrow + k) : (v2f){0.f, 0.f};
            acc = __builtin_amdgcn_wmma_f32_16x16x4_f32(
                false, a, false, b, (short)0, acc, false, false);
        }
    }
    {
        const float* wrow = Wr + (size_t)(colValid ? col : 0) * K;
        const float* arow;
        if constexpr (USE_LDS)
            arow = ldsX + (rowTileInBlk * 16 + l16) * LSTR;
        else
            arow = x + (size_t)(rowBase + l16) * K;  // nrows%16==0 -> in range
#pragma unroll 4
        for (int k0 = 0; k0 < K; k0 += 4) {
            int k = k0 + 2 * lh;
            v2f a = *(const v2f*)(arow + k);
            v2f b = colValid ? *(const v2f*)(wrow + k) : (v2f){0.f, 0.f};
            acc = __builtin_amdgcn_wmma_f32_16x16x4_f32(
                false, a, false, b, (short)0, acc, false, false);
        }
    }

    // Epilogue: nrows % 16 == 0 and the wave-uniform rowBase check above
    // guarantee all 16 tile rows are in range -> no per-element row guard.
    // Base pointer computed once; the 8 stores become immediate-offset
    // global_store_b32 (consecutive v -> +NOUT*4 bytes).
    const float bv = colValid ? bias[col] : 0.0f;
    float* po = out + (size_t)(rowBase + 8 * lh) * NOUT + col;
    if (colValid) {
#pragma unroll
        for (int v = 0; v < 8; ++v) {
            float val = acc[v] + bv;
            if (RELU) val = fmaxf(val, 0.0f);
            po[(size_t)v * NOUT] = val;
        }
    }
}

// ---------------------------------------------------------------------------
// Launch
// ---------------------------------------------------------------------------
extern "C" void kernel_launch(void* const* d_in, const int* in_sizes, int n_in,
                              void* d_out, int out_size, void* d_ws,
                              size_t ws_size, hipStream_t stream) {
    const float* x = (const float*)d_in[0];
    const long long* edge = (const long long*)d_in[1];  // int64 in reference
    const long long* esrc = edge;             // edge_index[0]
    const long long* edst = edge + N_EDGES;   // edge_index[1]
    const float* Wl1 = (const float*)d_in[2];
    const float* Wr1 = (const float*)d_in[3];
    const float* b1 = (const float*)d_in[4];
    const float* Wl2 = (const float*)d_in[5];
    const float* Wr2 = (const float*)d_in[6];
    const float* b2 = (const float*)d_in[7];
    const float* Wl3 = (const float*)d_in[8];
    const float* Wr3 = (const float*)d_in[9];
    const float* b3 = (const float*)d_in[10];
    const float* Wh = (const float*)d_in[11];
    const float* bh = (const float*)d_in[12];
    float* out = (float*)d_out;

    const int N = N_NODES, E = N_EDGES;
    float* cnt = (float*)d_ws;              // [N]
    float* inv = cnt + N;                   // [N]
    float* s = inv + N;                     // [N,128]
    float* bufA = s + (size_t)N * HIDDEN;   // [N,128]
    float* bufB = bufA + (size_t)N * HIDDEN;

    const int TB = 256;
    auto blks = [](long long n, int tb) { return (int)((n + tb - 1) / tb); };
    const int rowBlocks = N / 16;           // 3125 (exact)
    const int headBlocks = (rowBlocks + 7) / 8;

    // Degree counts (shared by all layers)
    zero_kernel<<<blks(N, TB), TB, 0, stream>>>(cnt, N);
    count_kernel<<<blks(E, TB), TB, 0, stream>>>(edst, cnt, E);
    inv_kernel<<<blks(N, TB), TB, 0, stream>>>(cnt, inv, N);

    // ---- Layer 1: IN_DIM=16 -> HIDDEN=128
    zero_kernel<<<blks((long long)N * IN_DIM, TB), TB, 0, stream>>>(
        s, (long long)N * IN_DIM);
    scatter_add_kernel<IN_DIM>
        <<<blks((long long)E * (IN_DIM / 4), TB), TB, 0, stream>>>(
            x, esrc, edst, s, E);
    sage_gemm_kernel<IN_DIM, HIDDEN, true, true>
        <<<rowBlocks, TB, 0, stream>>>(s, inv, x, Wl1, Wr1, b1, bufA, N);

    // ---- Layer 2: 128 -> 128
    zero_kernel<<<blks((long long)N * HIDDEN, TB), TB, 0, stream>>>(
        s, (long long)N * HIDDEN);
    scatter_add_kernel<HIDDEN>
        <<<blks((long long)E * (HIDDEN / 4), TB), TB, 0, stream>>>(
            bufA, esrc, edst, s, E);
    sage_gemm_kernel<HIDDEN, HIDDEN, true, true>
        <<<rowBlocks, TB, 0, stream>>>(s, inv, bufA, Wl2, Wr2, b2, bufB, N);

    // ---- Layer 3: 128 -> 128
    zero_kernel<<<blks((long long)N * HIDDEN, TB), TB, 0, stream>>>(
        s, (long long)N * HIDDEN);
    scatter_add_kernel<HIDDEN>
        <<<blks((long long)E * (HIDDEN / 4), TB), TB, 0, stream>>>(
            bufB, esrc, edst, s, E);
    sage_gemm_kernel<HIDDEN, HIDDEN, true, true>
        <<<rowBlocks, TB, 0, stream>>>(s, inv, bufB, Wl3, Wr3, b3, bufA, N);

    // ---- Head: 128 -> 4 (no aggregation term, no relu)
    sage_gemm_kernel<HIDDEN, OUT_DIM, false, false>
        <<<headBlocks, TB, 0, stream>>>(nullptr, nullptr, bufA, nullptr, Wh,
                                        bh, out, N);
}